// MincutPool_77747497992370
// MI455X (gfx1250) — compile-verified
//
#include <hip/hip_runtime.h>
#include <math.h>

#define B_    8192
#define NPG   40
#define DEG   16
#define IN_F  64
#define HID   32
#define OUTF  10
#define K1    20
#define K2    10
#define EPSF  1e-15f
#define NT    128   // 4 waves of 32

typedef __attribute__((ext_vector_type(2))) float v2f;
typedef __attribute__((ext_vector_type(8))) float v8f;

__global__ void zero_loss_kernel(float* loss) { *loss = 0.0f; }

__global__ void __launch_bounds__(NT) mincut_graph_kernel(
    const float* __restrict__ x,  const int* __restrict__ ei,
    const float* __restrict__ w_c1, const float* __restrict__ b_c1,
    const float* __restrict__ w_p1, const float* __restrict__ b_p1,
    const float* __restrict__ w2r,  const float* __restrict__ b2,
    const float* __restrict__ w2o,
    const float* __restrict__ w_p2, const float* __restrict__ b_p2,
    const float* __restrict__ w3r,  const float* __restrict__ b3,
    const float* __restrict__ w3o,
    const float* __restrict__ w_l1, const float* __restrict__ b_l1,
    const float* __restrict__ w_l2, const float* __restrict__ b_l2,
    float* __restrict__ out, float* __restrict__ loss_out)
{
    const int g    = blockIdx.x;
    const int tid  = threadIdx.x;
    const int lane = tid & 31;
    const int wave = tid >> 5;

    __shared__ __align__(16) float xs[48][IN_F]; // graph features, rows 40..47 zero-pad
    __shared__ __align__(16) float wc1[IN_F][HID];
    __shared__ float h[48][HID];        // x @ w_c1 (pre-normalization)
    __shared__ float hx[NPG][HID];      // GCN output (relu)
    __shared__ float adj[NPG][NPG];
    __shared__ float dinv[NPG];
    __shared__ float s1[NPG][K1];
    __shared__ float t1[NPG][K1];       // adj @ s1 ; later reused for ss
    __shared__ float xp[K1][HID];
    __shared__ float oadj[K1][K1];
    __shared__ float t2[K1][HID];       // scratch for dense conv (both stages)
    __shared__ float xn[K1][HID];
    __shared__ float s2[K1][K2];
    __shared__ float t3[K1][K2];        // oadj @ s2 ; later reused for ss2
    __shared__ float xp2[K2][HID];
    __shared__ float oadj2[K2][K2];
    __shared__ float xn3[K2][HID];
    __shared__ float dr1[K1];
    __shared__ float dr2[K2];
    __shared__ float pl[HID];
    __shared__ float zz[HID];
    __shared__ float red[8];            // 0:num1 1:den1 2:ssfro1 3:osum1 4:num2 5:den2 6:ssfro2 7:osum2

    // ---- vectorized loads: x tile (float4), w_c1 (float4); clear pads/adj/red ----
    {
        const float4* xv = (const float4*)(x + (size_t)g * NPG * IN_F);
        for (int i = tid; i < NPG * (IN_F / 4); i += NT) {       // 640 float4
            int r = i >> 4, c4 = i & 15;
            *(float4*)&xs[r][c4 * 4] = xv[i];
        }
        const float4 z4 = make_float4(0.f, 0.f, 0.f, 0.f);
        for (int i = tid; i < 8 * (IN_F / 4); i += NT) {         // pad rows 40..47
            int r = NPG + (i >> 4), c4 = i & 15;
            *(float4*)&xs[r][c4 * 4] = z4;
        }
        const float4* wv = (const float4*)w_c1;
        for (int i = tid; i < (IN_F * HID) / 4; i += NT)         // 512 float4
            ((float4*)&wc1[0][0])[i] = wv[i];
    }
    for (int i = tid; i < NPG * NPG; i += NT) (&adj[0][0])[i] = 0.0f;
    if (tid < 8) red[tid] = 0.0f;
    __syncthreads();

    // ---- to_dense_adj via LDS atomics (int4 edge loads; per-graph blocks are
    //      640 edges = multiple of 4, so 16B aligned) ----
    {
        const int eb = g * NPG * DEG;
        const int4* rv = (const int4*)(ei + eb);
        const int4* cv = (const int4*)(ei + (size_t)B_ * NPG * DEG + eb);
        const int base = g * NPG;
        for (int e4 = tid; e4 < (NPG * DEG) / 4; e4 += NT) {     // 160 int4 pairs
            int4 r4 = rv[e4];
            int4 c4 = cv[e4];
            atomicAdd(&adj[r4.x - base][c4.x - base], 1.0f);
            atomicAdd(&adj[r4.y - base][c4.y - base], 1.0f);
            atomicAdd(&adj[r4.z - base][c4.z - base], 1.0f);
            atomicAdd(&adj[r4.w - base][c4.w - base], 1.0f);
        }
    }

    // ---- h = xs @ w_c1 with V_WMMA_F32_16X16X4_F32 (48x64 * 64x32) ----
    // 6 output tiles (3 M-tiles x 2 N-tiles), distributed over 4 waves.
    {
        const int mlane = lane & 15;
        const int koff  = (lane >> 4) << 1;   // 0 or 2
        for (int t = wave; t < 6; t += 4) {
            const int mt = t >> 1, nt = t & 1;
            const int mrow = mt * 16 + mlane;
            const int ncol = nt * 16 + mlane;
            v8f acc = {};
            #pragma unroll
            for (int k0 = 0; k0 < IN_F; k0 += 4) {
                v2f a, b;
                a.x = xs[mrow][k0 + koff];
                a.y = xs[mrow][k0 + koff + 1];
                b.x = wc1[k0 + koff][ncol];
                b.y = wc1[k0 + koff + 1][ncol];
                acc = __builtin_amdgcn_wmma_f32_16x16x4_f32(
                        false, a, false, b, (short)0, acc, false, false);
            }
            const int rbase = mt * 16 + ((lane >> 4) << 3);
            #pragma unroll
            for (int v = 0; v < 8; ++v) h[rbase + v][ncol] = acc[v];
        }
    }
    __syncthreads();

    // ---- GCN: deg from column sums (+ self loop), dinv = rsqrt ----
    if (tid < NPG) {
        float d = 1.0f;
        for (int r = 0; r < NPG; ++r) d += adj[r][tid];
        dinv[tid] = rsqrtf(d);
    }
    __syncthreads();

    // out[c] = dinv[c]*sum_r adj[r][c]*dinv[r]*h[r] + dinv[c]^2*h[c] + b ; relu
    for (int i = tid; i < NPG * HID; i += NT) {
        int c = i >> 5, f = i & 31;
        float a = 0.0f;
        for (int r = 0; r < NPG; ++r) a += adj[r][c] * dinv[r] * h[r][f];
        a = dinv[c] * a + dinv[c] * dinv[c] * h[c][f] + b_c1[f];
        hx[c][f] = fmaxf(a, 0.0f);
    }
    __syncthreads();

    // ---- s1 = softmax(hx @ w_p1 + b_p1) per row ----
    if (tid < NPG) {
        float tmp[K1];
        float mx = -3.0e38f;
        #pragma unroll
        for (int k = 0; k < K1; ++k) {
            float a = b_p1[k];
            for (int c = 0; c < HID; ++c) a += hx[tid][c] * w_p1[c * K1 + k];
            tmp[k] = a; mx = fmaxf(mx, a);
        }
        float sum = 0.0f;
        #pragma unroll
        for (int k = 0; k < K1; ++k) { tmp[k] = __expf(tmp[k] - mx); sum += tmp[k]; }
        float inv = 1.0f / sum;
        #pragma unroll
        for (int k = 0; k < K1; ++k) s1[tid][k] = tmp[k] * inv;
    }
    __syncthreads();

    // ---- pool 1: xp = s1^T hx ; t1 = adj s1 ; oadj = s1^T t1 ----
    for (int i = tid; i < K1 * HID; i += NT) {
        int k = i >> 5, f = i & 31;
        float a = 0.0f;
        for (int n = 0; n < NPG; ++n) a += s1[n][k] * hx[n][f];
        xp[k][f] = a;
    }
    for (int i = tid; i < NPG * K1; i += NT) {
        int n = i / K1, l = i % K1;
        float a = 0.0f;
        for (int m = 0; m < NPG; ++m) a += adj[n][m] * s1[m][l];
        t1[n][l] = a;
    }
    __syncthreads();
    for (int i = tid; i < K1 * K1; i += NT) {
        int k = i / K1, l = i % K1;
        float a = 0.0f;
        for (int n = 0; n < NPG; ++n) a += s1[n][k] * t1[n][l];
        oadj[k][l] = a;
    }
    __syncthreads();

    // mincut_num1 = trace ; mincut_den1 = sum_n d_flat[n]*sum_k s1[n][k]^2
    {
        if (tid < K1) atomicAdd(&red[0], oadj[tid][tid]);
        if (tid < NPG) {
            float ds = 0.0f;
            for (int m = 0; m < NPG; ++m) ds += adj[tid][m];
            float sq = 0.0f;
            #pragma unroll
            for (int k = 0; k < K1; ++k) sq += s1[tid][k] * s1[tid][k];
            atomicAdd(&red[1], ds * sq);
        }
    }
    // ss1 = s1^T s1 (reuse t1 storage), Frobenius norm, ortho loss
    float* ss1 = &t1[0][0];
    {
        float p = 0.0f;
        for (int i = tid; i < K1 * K1; i += NT) {
            int k = i / K1, l = i % K1;
            float a = 0.0f;
            for (int n = 0; n < NPG; ++n) a += s1[n][k] * s1[n][l];
            ss1[i] = a; p += a * a;
        }
        atomicAdd(&red[2], p);
    }
    __syncthreads();
    {
        const float inrm = 1.0f / sqrtf(red[2]);
        const float dg = rsqrtf((float)K1);
        float p = 0.0f;
        for (int i = tid; i < K1 * K1; i += NT) {
            int k = i / K1, l = i % K1;
            float d = ss1[i] * inrm - ((k == l) ? dg : 0.0f);
            p += d * d;
        }
        atomicAdd(&red[3], p);
    }
    // zero diag, degree-normalize oadj
    if (tid < K1) oadj[tid][tid] = 0.0f;
    __syncthreads();
    if (tid < K1) {
        float rs = 0.0f;
        #pragma unroll
        for (int l = 0; l < K1; ++l) rs += oadj[tid][l];
        dr1[tid] = sqrtf(rs) + EPSF;
    }
    __syncthreads();
    for (int i = tid; i < K1 * K1; i += NT) {
        int k = i / K1, l = i % K1;
        oadj[k][l] = oadj[k][l] / (dr1[k] * dr1[l]);
    }
    __syncthreads();

    // ---- DenseGraphConv 2: xn = relu((oadj@xp)@w2r + b2 + xp@w2o) ----
    for (int i = tid; i < K1 * HID; i += NT) {
        int n = i >> 5, f = i & 31;
        float a = 0.0f;
        #pragma unroll
        for (int m = 0; m < K1; ++m) a += oadj[n][m] * xp[m][f];
        t2[n][f] = a;
    }
    __syncthreads();
    for (int i = tid; i < K1 * HID; i += NT) {
        int n = i >> 5, f = i & 31;
        float a = b2[f];
        for (int c = 0; c < HID; ++c)
            a += t2[n][c] * w2r[c * HID + f] + xp[n][c] * w2o[c * HID + f];
        xn[n][f] = fmaxf(a, 0.0f);
    }
    __syncthreads();

    // ---- s2 = softmax(xn @ w_p2 + b_p2) ----
    if (tid < K1) {
        float tmp[K2];
        float mx = -3.0e38f;
        #pragma unroll
        for (int k = 0; k < K2; ++k) {
            float a = b_p2[k];
            for (int c = 0; c < HID; ++c) a += xn[tid][c] * w_p2[c * K2 + k];
            tmp[k] = a; mx = fmaxf(mx, a);
        }
        float sum = 0.0f;
        #pragma unroll
        for (int k = 0; k < K2; ++k) { tmp[k] = __expf(tmp[k] - mx); sum += tmp[k]; }
        float inv = 1.0f / sum;
        #pragma unroll
        for (int k = 0; k < K2; ++k) s2[tid][k] = tmp[k] * inv;
    }
    __syncthreads();

    // ---- pool 2 ----
    for (int i = tid; i < K2 * HID; i += NT) {
        int k = i >> 5, f = i & 31;
        float a = 0.0f;
        #pragma unroll
        for (int n = 0; n < K1; ++n) a += s2[n][k] * xn[n][f];
        xp2[k][f] = a;
    }
    for (int i = tid; i < K1 * K2; i += NT) {
        int n = i / K2, l = i % K2;
        float a = 0.0f;
        #pragma unroll
        for (int m = 0; m < K1; ++m) a += oadj[n][m] * s2[m][l];
        t3[n][l] = a;
    }
    __syncthreads();
    for (int i = tid; i < K2 * K2; i += NT) {
        int k = i / K2, l = i % K2;
        float a = 0.0f;
        #pragma unroll
        for (int n = 0; n < K1; ++n) a += s2[n][k] * t3[n][l];
        oadj2[k][l] = a;
    }
    __syncthreads();
    {
        if (tid < K2) atomicAdd(&red[4], oadj2[tid][tid]);
        if (tid < K1) {
            float ds = 0.0f;
            #pragma unroll
            for (int m = 0; m < K1; ++m) ds += oadj[tid][m];
            float sq = 0.0f;
            #pragma unroll
            for (int k = 0; k < K2; ++k) sq += s2[tid][k] * s2[tid][k];
            atomicAdd(&red[5], ds * sq);
        }
    }
    float* ss2 = &t3[0][0];   // t3 free now; 100 <= 200 slots
    {
        float p = 0.0f;
        for (int i = tid; i < K2 * K2; i += NT) {
            int k = i / K2, l = i % K2;
            float a = 0.0f;
            #pragma unroll
            for (int n = 0; n < K1; ++n) a += s2[n][k] * s2[n][l];
            ss2[i] = a; p += a * a;
        }
        if (tid < K2 * K2) atomicAdd(&red[6], p);
    }
    __syncthreads();
    {
        const float inrm = 1.0f / sqrtf(red[6]);
        const float dg = rsqrtf((float)K2);
        float p = 0.0f;
        for (int i = tid; i < K2 * K2; i += NT) {
            int k = i / K2, l = i % K2;
            float d = ss2[i] * inrm - ((k == l) ? dg : 0.0f);
            p += d * d;
        }
        if (tid < K2 * K2) atomicAdd(&red[7], p);
    }
    if (tid < K2) oadj2[tid][tid] = 0.0f;
    __syncthreads();
    if (tid < K2) {
        float rs = 0.0f;
        #pragma unroll
        for (int l = 0; l < K2; ++l) rs += oadj2[tid][l];
        dr2[tid] = sqrtf(rs) + EPSF;
    }
    __syncthreads();
    for (int i = tid; i < K2 * K2; i += NT) {
        int k = i / K2, l = i % K2;
        oadj2[k][l] = oadj2[k][l] / (dr2[k] * dr2[l]);
    }
    __syncthreads();

    // ---- DenseGraphConv 3 (no relu) ----
    for (int i = tid; i < K2 * HID; i += NT) {
        int n = i >> 5, f = i & 31;
        float a = 0.0f;
        #pragma unroll
        for (int m = 0; m < K2; ++m) a += oadj2[n][m] * xp2[m][f];
        t2[n][f] = a;
    }
    __syncthreads();
    for (int i = tid; i < K2 * HID; i += NT) {
        int n = i >> 5, f = i & 31;
        float a = b3[f];
        for (int c = 0; c < HID; ++c)
            a += t2[n][c] * w3r[c * HID + f] + xp2[n][c] * w3o[c * HID + f];
        xn3[n][f] = a;
    }
    __syncthreads();

    // ---- mean pool + MLP head ----
    if (tid < HID) {
        float a = 0.0f;
        #pragma unroll
        for (int n = 0; n < K2; ++n) a += xn3[n][tid];
        pl[tid] = a * (1.0f / (float)K2);
    }
    __syncthreads();
    if (tid < HID) {
        float a = b_l1[tid];
        for (int c = 0; c < HID; ++c) a += pl[c] * w_l1[c * HID + tid];
        zz[tid] = fmaxf(a, 0.0f);
    }
    __syncthreads();
    if (tid < OUTF) {
        float a = b_l2[tid];
        for (int c = 0; c < HID; ++c) a += zz[c] * w_l2[c * OUTF + tid];
        out[(size_t)g * OUTF + tid] = a;
    }

    // ---- loss contribution: (mc1 + mc2 + o1 + o2) / B ----
    if (tid == 0) {
        float contrib = (-(red[0] / red[1]) - (red[4] / red[5])
                         + sqrtf(red[3]) + sqrtf(red[7])) * (1.0f / (float)B_);
        atomicAdd(loss_out, contrib);
    }
}

extern "C" void kernel_launch(void* const* d_in, const int* in_sizes, int n_in,
                              void* d_out, int out_size, void* d_ws, size_t ws_size,
                              hipStream_t stream) {
    const float* x    = (const float*)d_in[0];
    const int*   ei   = (const int*)  d_in[1];
    // d_in[2] = batch (structure is implicit; unused)
    const float* w_c1 = (const float*)d_in[3];
    const float* b_c1 = (const float*)d_in[4];
    const float* w_p1 = (const float*)d_in[5];
    const float* b_p1 = (const float*)d_in[6];
    const float* w2r  = (const float*)d_in[7];
    const float* b2   = (const float*)d_in[8];
    const float* w2o  = (const float*)d_in[9];
    const float* w_p2 = (const float*)d_in[10];
    const float* b_p2 = (const float*)d_in[11];
    const float* w3r  = (const float*)d_in[12];
    const float* b3   = (const float*)d_in[13];
    const float* w3o  = (const float*)d_in[14];
    const float* w_l1 = (const float*)d_in[15];
    const float* b_l1 = (const float*)d_in[16];
    const float* w_l2 = (const float*)d_in[17];
    const float* b_l2 = (const float*)d_in[18];
    float* out = (float*)d_out;

    zero_loss_kernel<<<1, 1, 0, stream>>>(out + (size_t)B_ * OUTF);
    mincut_graph_kernel<<<B_, NT, 0, stream>>>(
        x, ei, w_c1, b_c1, w_p1, b_p1, w2r, b2, w2o,
        w_p2, b_p2, w3r, b3, w3o, w_l1, b_l1, w_l2, b_l2,
        out, out + (size_t)B_ * OUTF);
}